// Map3D_77137612636268
// MI455X (gfx1250) — compile-verified
//
#include <hip/hip_runtime.h>

// ---------------- problem constants (from reference) ----------------
#define SZ      48
#define DE      44
#define BATCH   4
#define FLATN   (DE*SZ*SZ)        // 101376
#define NVOX    (BATCH*FLATN)     // 405504
#define CONVEY  0.9f
#define H1      1024
#define H2      128
#define NOUT    10

// split-K GEMM1 config
#define KSPLIT  64
#define KCHUNK  (FLATN/KSPLIT)    // 1584 (divisible by 4)
#define NT1     (H1/16)           // 64 N-tiles
#define NTPB    8                 // N-tiles (waves) per block in gemm1

typedef __attribute__((ext_vector_type(2))) float v2f;
typedef __attribute__((ext_vector_type(4))) float v4f;
typedef __attribute__((ext_vector_type(8))) float v8f;

// ---------------- neu0 = vinput + frame ----------------
__global__ __launch_bounds__(256) void init_neu(const float* __restrict__ vin,
                                                const float* __restrict__ frame,
                                                float* __restrict__ neu) {
    int v = blockIdx.x * 256 + threadIdx.x;
    if (v >= NVOX) return;
    int f = v % FLATN;
    neu[v] = vin[v] + frame[f];
}

// ---------------- locally-connected 6x6x6 step ----------------
// out[b,i,j,k] = relu( sum_{di,dj,dk} neu[b,i+di-3,j+dj-3,k+dk-3] * syn[f,di,dj,dk] ) * 0.9
// Per-voxel synapse row = 216 contiguous floats (864 B, 16B-aligned). Each di-block is
// 36 floats = 144 B = nine aligned float4 -> global_load_b128 streaming; L2-resident
// (87.6 MB < 192 MB L2) for iterations 2..5.
__global__ __launch_bounds__(256) void local_step(const float* __restrict__ nin,
                                                  const float* __restrict__ syn,
                                                  float* __restrict__ nout) {
    int v = blockIdx.x * 256 + threadIdx.x;
    if (v >= NVOX) return;
    int b = v / FLATN;
    int f = v - b * FLATN;
    int i = f / (SZ * SZ);
    int rem = f - i * SZ * SZ;
    int j = rem / SZ;
    int k = rem - j * SZ;

    const float* __restrict__ s  = syn + (size_t)f * 216;
    const float* __restrict__ nb = nin + (size_t)b * FLATN;

    float acc = 0.f;
    for (int di = 0; di < 6; ++di) {
        // 36 weights for this di slab as 9 aligned vec4 loads (b128)
        const v4f* __restrict__ s4 = (const v4f*)(s + di * 36);
        v4f sv[9];
#pragma unroll
        for (int q = 0; q < 9; ++q) sv[q] = s4[q];
        const float* w = (const float*)sv;   // compile-time indexed after unroll

        int ii = i + di - 3;
        if ((unsigned)ii >= (unsigned)DE) continue;   // padded rows contribute 0
        const float* __restrict__ np = nb + ii * SZ * SZ;
#pragma unroll
        for (int dj = 0; dj < 6; ++dj) {
            int jj = j + dj - 3;
            if ((unsigned)jj >= (unsigned)SZ) continue;
            const float* __restrict__ nr = np + jj * SZ;
#pragma unroll
            for (int dk = 0; dk < 6; ++dk) {
                int kk = k + dk - 3;
                if ((unsigned)kk < (unsigned)SZ)
                    acc = fmaf(nr[kk], w[dj * 6 + dk], acc);
            }
        }
    }
    nout[v] = fmaxf(acc, 0.f) * CONVEY;
}

// ---------------- pack final neu into paired, M-padded A: xP[k/2][16][2] ----------------
// Lane's two A operand values (k, k+1) become adjacent -> global_load_b64 in gemm1.
__global__ __launch_bounds__(256) void pack_xP(const float* __restrict__ neu,
                                               float* __restrict__ xP) {
    int t = blockIdx.x * 256 + threadIdx.x;   // 0 .. FLATN/2*16 - 1
    int p = t >> 4, m = t & 15;               // k-pair index, padded batch row
    int k = p * 2;
    v2f val;
    val.x = (m < BATCH) ? neu[(size_t)m * FLATN + k]     : 0.f;
    val.y = (m < BATCH) ? neu[(size_t)m * FLATN + k + 1] : 0.f;
    *(v2f*)(xP + (size_t)p * 32 + m * 2) = val;
}

// ---------------- GEMM1: split-K, 8 waves (8 N-tiles) per block ----------------
// All 8 waves share one A stream (same K-split) -> WGP$ hits; their B slices cover
// 512 contiguous bytes per W1 row -> full cache-line utilization of the 415 MB stream.
__global__ __launch_bounds__(256) void gemm1(const float* __restrict__ xP,
                                             const float* __restrict__ W1,
                                             float* __restrict__ part) {
    int lane = threadIdx.x & 31;            // EXEC all ones per wave (WMMA requirement)
    int wv   = threadIdx.x >> 5;            // 0..7
    int nt   = blockIdx.x * NTPB + wv;      // N tile 0..63
    int sp   = blockIdx.y;                  // K split 0..63
    int lm   = lane & 15;
    int koff = (lane >> 4) ? 2 : 0;         // ISA 16x4 f32 A layout: high half-wave = K+2
    int k0   = sp * KCHUNK;

    const float* __restrict__ xp = xP + (size_t)k0 * 16 + lm * 2;  // paired layout
    const float* __restrict__ wp = W1 + (size_t)k0 * H1 + nt * 16 + lm;

    v8f acc = {};
    for (int kk = 0; kk < KCHUNK; kk += 4) {
        int kb = kk + koff;
        v2f a = *(const v2f*)(xp + (size_t)(kb >> 1) * 32);        // b64 load
        v2f b;
        b.x = wp[(size_t)kb * H1];
        b.y = wp[(size_t)(kb + 1) * H1];
        acc = __builtin_amdgcn_wmma_f32_16x16x4_f32(
                  false, a, false, b, (short)0, acc, false, false);
    }
    float* __restrict__ p = part + ((size_t)sp * NT1 + nt) * 256;
#pragma unroll
    for (int r = 0; r < 8; ++r) p[r * 32 + lane] = acc[r];
}

// ---------------- reduce split-K partials, bias + relu, store h1 paired: h1P[k/2][16][2] ----------------
__global__ __launch_bounds__(256) void reduce1(const float* __restrict__ part,
                                               const float* __restrict__ b1,
                                               float* __restrict__ h1P) {
    int nt = blockIdx.x;                    // 0..63
    int e  = threadIdx.x;                   // 0..255 tile element (r*32+lane)
    int r = e >> 5, lane = e & 31;
    float sum = 0.f;
    const float* __restrict__ p = part + (size_t)nt * 256 + e;
    for (int sp = 0; sp < KSPLIT; ++sp)     // fixed order -> deterministic
        sum += p[(size_t)sp * NT1 * 256];
    int M = r + ((lane >> 4) ? 8 : 0);      // documented C/D layout
    int N = nt * 16 + (lane & 15);          // k-dim of layer 2
    h1P[(N >> 1) * 32 + M * 2 + (N & 1)] = fmaxf(sum + b1[N], 0.f);
}

// ---------------- GEMM2: (16x1024) @ W2(1024x128), 8 waves in one block ----------------
__global__ __launch_bounds__(256) void gemm2(const float* __restrict__ h1P,
                                             const float* __restrict__ W2,
                                             const float* __restrict__ b2,
                                             float* __restrict__ h2T) {
    int lane = threadIdx.x & 31;
    int nt   = threadIdx.x >> 5;            // 0..7
    int lm   = lane & 15;
    int koff = (lane >> 4) ? 2 : 0;

    const float* __restrict__ xp = h1P + lm * 2;
    const float* __restrict__ wp = W2 + nt * 16 + lm;

    v8f acc = {};
    for (int kk = 0; kk < H1; kk += 4) {
        int kb = kk + koff;
        v2f a = *(const v2f*)(xp + (kb >> 1) * 32);
        v2f b;
        b.x = wp[kb * H2];
        b.y = wp[(kb + 1) * H2];
        acc = __builtin_amdgcn_wmma_f32_16x16x4_f32(
                  false, a, false, b, (short)0, acc, false, false);
    }
    int Mb = (lane >> 4) ? 8 : 0;
    int N  = nt * 16 + lm;
#pragma unroll
    for (int r = 0; r < 8; ++r)
        h2T[N * 16 + (Mb + r)] = fmaxf(acc[r] + b2[N], 0.f);
}

// ---------------- layer 3: (4x128) @ (128x10) + b3 ----------------
__global__ __launch_bounds__(64) void layer3(const float* __restrict__ h2T,
                                             const float* __restrict__ W3,
                                             const float* __restrict__ b3,
                                             float* __restrict__ out) {
    int t = threadIdx.x;
    if (t >= BATCH * NOUT) return;
    int b = t / NOUT, o = t - b * NOUT;
    float sum = b3[o];
    for (int c = 0; c < H2; ++c)
        sum = fmaf(h2T[c * 16 + b], W3[c * NOUT + o], sum);
    out[t] = sum;
}

// ---------------- host-side orchestration ----------------
extern "C" void kernel_launch(void* const* d_in, const int* in_sizes, int n_in,
                              void* d_out, int out_size, void* d_ws, size_t ws_size,
                              hipStream_t stream) {
    const float* vin   = (const float*)d_in[0];
    const float* frame = (const float*)d_in[1];
    const float* syn   = (const float*)d_in[2];
    const float* W1    = (const float*)d_in[3];
    const float* b1    = (const float*)d_in[4];
    const float* W2    = (const float*)d_in[5];
    const float* b2    = (const float*)d_in[6];
    const float* W3    = (const float*)d_in[7];
    const float* b3    = (const float*)d_in[8];
    float* out = (float*)d_out;

    // workspace layout (bytes, 256B-aligned)
    char* ws = (char*)d_ws;
    float* neuA = (float*)(ws);                                  // 1,622,016 B
    float* neuB = (float*)(ws + 1622016);                        // 1,622,016 B
    float* xP   = (float*)(ws + 3244032);                        // 6,488,064 B
    float* part = (float*)(ws + 9732096);                        // 4,194,304 B
    float* h1P  = (float*)(ws + 13926400);                       //    65,536 B
    float* h2T  = (float*)(ws + 13991936);                       //     8,192 B

    const int vb = NVOX / 256;   // 1584 exactly

    init_neu<<<vb, 256, 0, stream>>>(vin, frame, neuA);

    // 5 locally-connected steps, ping-pong A->B->A->B->A->B (final in neuB)
    local_step<<<vb, 256, 0, stream>>>(neuA, syn, neuB);
    local_step<<<vb, 256, 0, stream>>>(neuB, syn, neuA);
    local_step<<<vb, 256, 0, stream>>>(neuA, syn, neuB);
    local_step<<<vb, 256, 0, stream>>>(neuB, syn, neuA);
    local_step<<<vb, 256, 0, stream>>>(neuA, syn, neuB);

    pack_xP<<<(FLATN / 2 * 16) / 256, 256, 0, stream>>>(neuB, xP);   // 3168 blocks

    gemm1<<<dim3(NT1 / NTPB, KSPLIT), 256, 0, stream>>>(xP, W1, part);
    reduce1<<<NT1, 256, 0, stream>>>(part, b1, h1P);
    gemm2<<<1, 256, 0, stream>>>(h1P, W2, b2, h2T);
    layer3<<<1, 64, 0, stream>>>(h2T, W3, b3, out);
}